// GAT_78245714198780
// MI455X (gfx1250) — compile-verified
//
#include <hip/hip_runtime.h>
#include <hip/hip_bf16.h>
#include <math.h>

#define GN 100000   // nodes
#define GE 800000   // edges
#define GH 4        // heads
#define SLOPE 0.2f

typedef float v2f __attribute__((ext_vector_type(2)));
typedef float v8f __attribute__((ext_vector_type(8)));

// ---------------------------------------------------------------------------
// float atomic max via sign-bucketed integer atomics (order-independent,
// deterministic; works for any sign given init = -inf)
// ---------------------------------------------------------------------------
__device__ __forceinline__ void atomicMaxFloat(float* addr, float val) {
  if (val >= 0.0f) {
    atomicMax((int*)addr, __float_as_int(val));
  } else {
    atomicMin((unsigned int*)addr, __float_as_uint(val));
  }
}

// ---------------------------------------------------------------------------
// fp32 WMMA GEMM: C[M,Nc] = A[M,K] * B[K,Nc], row-major.
// Each wave computes a 16x32 output slab (two 16x16 WMMA tiles sharing the
// same A fragment -> halves A traffic, doubles WMMA density per byte).
// A layout (16x4 f32): lane l -> m = l&15, VGPR j holds K = (l>>4)*2 + j
// B layout (4x16 f32): lane l -> n = l&15, VGPR v holds K = (l>>4)*2 + v
// C layout (16x16 f32): lane l -> n = l&15, VGPR v holds m = v + (l>>4)*8
// ---------------------------------------------------------------------------
__global__ void __launch_bounds__(256)
k_gemm_wmma(const float* __restrict__ A, const float* __restrict__ B,
            float* __restrict__ C, int K, int Nc, int pairsPerRow,
            int numPairs) {
  int wave = (int)((blockIdx.x * blockDim.x + threadIdx.x) >> 5);  // uniform
  int lane = threadIdx.x & 31;
  if (wave >= numPairs) return;  // whole wave exits together; EXEC stays full

  int mt = wave / pairsPerRow;
  int pt = wave - mt * pairsPerRow;
  int row0 = mt * 16, col0 = pt * 32;

  int nl   = lane & 15;   // m for A loads, n for B/C
  int kgrp = lane >> 4;   // 0/1 -> K offset 0/2

  const float* __restrict__ ap  = A + (size_t)(row0 + nl) * K + kgrp * 2;
  const float* __restrict__ bp0 = B + (size_t)(kgrp * 2) * Nc + col0 + nl;
  const float* __restrict__ bp1 = bp0 + 16;
  const size_t bstep = (size_t)4 * Nc;

  v8f acc0 = {0.f, 0.f, 0.f, 0.f, 0.f, 0.f, 0.f, 0.f};
  v8f acc1 = {0.f, 0.f, 0.f, 0.f, 0.f, 0.f, 0.f, 0.f};

#pragma unroll 4
  for (int k = 0; k < K; k += 4) {
    v2f a = *(const v2f*)ap;            // A[m][k+2*kgrp .. +1], 8B aligned
    ap += 4;
    v2f b0, b1;
    b0.x = bp0[0];  b0.y = bp0[Nc];     // B[k+2*kgrp(+1)][col0 + n]
    b1.x = bp1[0];  b1.y = bp1[Nc];     // B[k+2*kgrp(+1)][col0 + 16 + n]
    bp0 += bstep;
    bp1 += bstep;
    acc0 = __builtin_amdgcn_wmma_f32_16x16x4_f32(false, a, false, b0,
                                                 (short)0, acc0, false, false);
    acc1 = __builtin_amdgcn_wmma_f32_16x16x4_f32(false, a, false, b1,
                                                 (short)0, acc1, false, false);
  }

  float* cptr = C + (size_t)(row0 + kgrp * 8) * Nc + col0 + nl;
#pragma unroll
  for (int v = 0; v < 8; ++v) {
    cptr[(size_t)v * Nc]      = acc0[v];
    cptr[(size_t)v * Nc + 16] = acc1[v];
  }
}

// ---------------------------------------------------------------------------
// el/er attention coefficients + init emax/-inf and den/0. One thread = (n,h).
// feat is [N][H*Do]; slice (n,h) starts at i*Do where i = n*H+h.
// ---------------------------------------------------------------------------
__global__ void __launch_bounds__(256)
k_attn_prep(const float* __restrict__ feat, const float* __restrict__ al,
            const float* __restrict__ ar, float* __restrict__ el,
            float* __restrict__ er, float* __restrict__ emax,
            float* __restrict__ den, int Do, int total) {
  int i = blockIdx.x * blockDim.x + threadIdx.x;
  if (i >= total) return;
  int h = i & (GH - 1);
  const float* f  = feat + (size_t)i * Do;
  const float* pl = al + h * Do;
  const float* pr = ar + h * Do;
  float sl = 0.f, sr = 0.f;
#pragma unroll 8
  for (int d = 0; d < Do; ++d) {
    float v = f[d];
    sl += v * pl[d];
    sr += v * pr[d];
  }
  el[i] = sl;
  er[i] = sr;
  emax[i] = -__builtin_huge_valf();
  den[i]  = 0.f;
}

__global__ void __launch_bounds__(256)
k_zero(float* __restrict__ p, int total) {
  int i = blockIdx.x * blockDim.x + threadIdx.x;
  if (i < total) p[i] = 0.f;
}

// edge pass 1: segment max of leaky-relu logits (one thread per edge-head)
__global__ void __launch_bounds__(256)
k_edge_max(const int* __restrict__ src, const int* __restrict__ dst,
           const float* __restrict__ el, const float* __restrict__ er,
           float* __restrict__ emax, int totalEH) {
  int i = blockIdx.x * blockDim.x + threadIdx.x;
  if (i >= totalEH) return;
  int e = i >> 2, h = i & 3;
  int s = src[e], d = dst[e];
  float x = el[s * GH + h] + er[d * GH + h];
  x = x > 0.f ? x : SLOPE * x;
  atomicMaxFloat(emax + d * GH + h, x);
}

// edge pass 2: ee = exp(e - emax[dst]); den[dst] += ee
__global__ void __launch_bounds__(256)
k_edge_exp(const int* __restrict__ src, const int* __restrict__ dst,
           const float* __restrict__ el, const float* __restrict__ er,
           const float* __restrict__ emax, float* __restrict__ ee,
           float* __restrict__ den, int totalEH) {
  int i = blockIdx.x * blockDim.x + threadIdx.x;
  if (i >= totalEH) return;
  int e = i >> 2, h = i & 3;
  int s = src[e], d = dst[e];
  float x = el[s * GH + h] + er[d * GH + h];
  x = x > 0.f ? x : SLOPE * x;
  float v = expf(x - emax[d * GH + h]);
  ee[i] = v;
  atomicAdd(den + d * GH + h, v);
}

// edge aggregate: one wave per edge; msg[dst][:] += ee[e][h] * feat[src][:]
__global__ void __launch_bounds__(256)
k_edge_agg(const int* __restrict__ src, const int* __restrict__ dst,
           const float* __restrict__ ee, const float* __restrict__ feat,
           float* __restrict__ msg, int Do, int F, int nE) {
  int wave = (int)((blockIdx.x * blockDim.x + threadIdx.x) >> 5);
  int lane = threadIdx.x & 31;
  if (wave >= nE) return;
  int s = src[wave], d = dst[wave];
  const float* fs = feat + (size_t)s * F;
  float* md = msg + (size_t)d * F;
  const float* eb = ee + (size_t)wave * GH;
  float e0 = eb[0], e1 = eb[1], e2 = eb[2], e3 = eb[3];
  for (int j = lane; j < F; j += 32) {
    int h = j / Do;
    float w = (h == 0) ? e0 : (h == 1) ? e1 : (h == 2) ? e2 : e3;
    atomicAdd(md + j, w * fs[j]);
  }
}

// finalize layers 0/1: h = elu(msg/den), Do fixed at 64 (h = j>>6)
__global__ void __launch_bounds__(256)
k_fin_elu(const float* __restrict__ msg, const float* __restrict__ den,
          float* __restrict__ hout, int F, int total) {
  int i = blockIdx.x * blockDim.x + threadIdx.x;
  if (i >= total) return;
  int n = i / F;
  int j = i - n * F;
  int h = j >> 6;  // Do == 64 for layers 0 and 1
  float dn = den[n * GH + h];
  float v = dn > 0.f ? msg[i] / dn : 0.f;      // isolated node -> 0 (matches ref)
  hout[i] = v > 0.f ? v : expf(v) - 1.f;       // ELU
}

// finalize layer 2: logits = mean over heads of msg/den  (C = 40)
__global__ void __launch_bounds__(256)
k_fin_mean(const float* __restrict__ msg, const float* __restrict__ den,
           float* __restrict__ out, int total) {
  int i = blockIdx.x * blockDim.x + threadIdx.x;
  if (i >= total) return;
  int n = i / 40;
  int c = i - n * 40;
  float acc = 0.f;
#pragma unroll
  for (int h = 0; h < GH; ++h) {
    float dn = den[n * GH + h];
    acc += dn > 0.f ? msg[(size_t)n * 160 + h * 40 + c] / dn : 0.f;
  }
  out[i] = 0.25f * acc;
}

// ---------------------------------------------------------------------------
extern "C" void kernel_launch(void* const* d_in, const int* in_sizes, int n_in,
                              void* d_out, int out_size, void* d_ws, size_t ws_size,
                              hipStream_t stream) {
  const float* x   = (const float*)d_in[0];
  const float* W0  = (const float*)d_in[1];
  const float* al0 = (const float*)d_in[2];
  const float* ar0 = (const float*)d_in[3];
  const float* W1  = (const float*)d_in[4];
  const float* al1 = (const float*)d_in[5];
  const float* ar1 = (const float*)d_in[6];
  const float* W2  = (const float*)d_in[7];
  const float* al2 = (const float*)d_in[8];
  const float* ar2 = (const float*)d_in[9];
  const int*   src = (const int*)d_in[10];
  const int*   dst = (const int*)d_in[11];
  float* out = (float*)d_out;

  float* ws   = (float*)d_ws;
  float* buf0 = ws;                               // N*256  (feat)
  float* buf1 = buf0 + (size_t)GN * 256;          // N*256
  float* buf2 = buf1 + (size_t)GN * 256;          // N*256
  float* el   = buf2 + (size_t)GN * 256;          // N*H
  float* er   = el   + (size_t)GN * GH;
  float* emax = er   + (size_t)GN * GH;
  float* den  = emax + (size_t)GN * GH;
  float* ee   = den  + (size_t)GN * GH;           // E*H

  const int NH = GN * GH;      // 400000
  const int EH = GE * GH;      // 3200000
  auto cdiv = [](long a, long b) { return (int)((a + b - 1) / b); };

  // Runs GEMM + edge-softmax message passing; msg/den left for finalize.
  auto gat_core = [&](const float* hin, int K, const float* W, const float* al,
                      const float* ar, int Do, int Fout, float* feat, float* msg) {
    int pairsPerRow = Fout / 32;                   // 16x32 slab per wave
    int numPairs = (GN / 16) * pairsPerRow;
    k_gemm_wmma<<<cdiv(numPairs, 8), 256, 0, stream>>>(hin, W, feat, K, Fout,
                                                       pairsPerRow, numPairs);
    k_attn_prep<<<cdiv(NH, 256), 256, 0, stream>>>(feat, al, ar, el, er, emax,
                                                   den, Do, NH);
    k_zero<<<cdiv((long)GN * Fout, 256), 256, 0, stream>>>(msg, GN * Fout);
    k_edge_max<<<cdiv(EH, 256), 256, 0, stream>>>(src, dst, el, er, emax, EH);
    k_edge_exp<<<cdiv(EH, 256), 256, 0, stream>>>(src, dst, el, er, emax, ee,
                                                  den, EH);
    k_edge_agg<<<cdiv(GE, 8), 256, 0, stream>>>(src, dst, ee, feat, msg, Do,
                                                Fout, GE);
  };

  // Layer 0: x[N,128] -> buf1 = elu(gat)
  gat_core(x, 128, W0, al0, ar0, 64, 256, buf0, buf1);
  k_fin_elu<<<cdiv((long)GN * 256, 256), 256, 0, stream>>>(buf1, den, buf1, 256,
                                                           GN * 256);
  // Layer 1: buf1[N,256] -> buf2 = elu(gat)
  gat_core(buf1, 256, W1, al1, ar1, 64, 256, buf0, buf2);
  k_fin_elu<<<cdiv((long)GN * 256, 256), 256, 0, stream>>>(buf2, den, buf2, 256,
                                                           GN * 256);
  // Layer 2: buf2[N,256] -> out[N,40] = mean over heads
  gat_core(buf2, 256, W2, al2, ar2, 40, 160, buf0, buf1);
  k_fin_mean<<<cdiv((long)GN * 40, 256), 256, 0, stream>>>(buf1, den, out,
                                                           GN * 40);
}